// LearnableGate16Array_33913061769292
// MI455X (gfx1250) — compile-verified
//
#include <hip/hip_runtime.h>

#define GATES 32768
#define BATCH 512

typedef float v2f __attribute__((ext_vector_type(2)));
typedef float v4f __attribute__((ext_vector_type(4)));
typedef float v8f __attribute__((ext_vector_type(8)));

// Coefficient matrix for the 16 gates expressed in the {1, A, B, A*B} basis.
// Row m = which fused weight (0:const, 1:A, 2:B, 3:AB), col k = gate index.
// Rows 4..15 are zero (M padded to 16 for the 16x16x4 WMMA).
__device__ __constant__ float kCoef[16][16] = {
    //        k: 0  1   2  3   4  5   6   7   8   9  10  11  12  13  14 15
    /*const*/ { 0, 0,  0, 0,  0, 0,  0,  0,  1,  1,  1,  1,  1,  1,  1, 1},
    /*A    */ { 0, 0,  1, 1,  0, 0,  1,  1, -1, -1,  0,  0, -1, -1,  0, 0},
    /*B    */ { 0, 0,  0, 0,  1, 1,  1,  1, -1, -1, -1, -1,  0,  0,  0, 0},
    /*AB   */ { 0, 1, -1, 0, -1, 0, -2, -1,  1,  2,  0,  1,  0,  1, -1, 0},
    // remaining rows implicitly zero
};

// ---------------------------------------------------------------------------
// Kernel 1: per gate column g, softmax over the 16 logits, then fuse the 16
// gate weights into 4 coefficients W0..W3 via Coef(16x16) x E(16x16-cols)
// using chained V_WMMA_F32_16X16X4_F32 (K = 4*4 = 16).
//
// Wave layout: each wave handles 16 columns. Lane L and lane L+16 both own
// column base+(L&15) (redundant softmax) so that both half-waves have the
// exp values locally for the B-operand striping:
//   chunk c: B vgpr0 = row 4c   (lanes 0-15) / row 4c+2 (lanes 16-31)
//            B vgpr1 = row 4c+1 (lanes 0-15) / row 4c+3 (lanes 16-31)
// A operand follows the documented 16x4 f32 layout (same k split).
// D: vgpr v holds row v for lanes 0-15 -> lane L<16 has W0..W3 of its column
// in acc[0..3].
// ---------------------------------------------------------------------------
__global__ void lg16_prep_weights(const float* __restrict__ w,
                                  float* __restrict__ Wfused) {
  const int lane = threadIdx.x & 31;
  const int wave = threadIdx.x >> 5;
  const int colBase = (blockIdx.x * 8 + wave) * 16;
  const int col = colBase + (lane & 15);
  const bool hi = (threadIdx.x & 16) != 0;

  // Load 16 logits for this column, softmax (unnormalized exps; divide at end)
  float e[16];
#pragma unroll
  for (int k = 0; k < 16; ++k) e[k] = w[k * GATES + col];
  float mx = e[0];
#pragma unroll
  for (int k = 1; k < 16; ++k) mx = fmaxf(mx, e[k]);
  float s = 0.0f;
#pragma unroll
  for (int k = 0; k < 16; ++k) {
    e[k] = __expf(e[k] - mx);
    s += e[k];
  }

  const int m = lane & 15;
  v8f acc = {};  // 16x16 f32 accumulator, starts at zero
#pragma unroll
  for (int c = 0; c < 4; ++c) {
    const int kbase = 4 * c + (hi ? 2 : 0);
    v2f a;
    a[0] = kCoef[m][kbase];
    a[1] = kCoef[m][kbase + 1];
    v2f b;
    b[0] = hi ? e[4 * c + 2] : e[4 * c + 0];
    b[1] = hi ? e[4 * c + 3] : e[4 * c + 1];
    acc = __builtin_amdgcn_wmma_f32_16x16x4_f32(
        /*neg_a=*/false, a, /*neg_b=*/false, b,
        /*c_mod=*/(short)0, acc, /*reuse_a=*/false, /*reuse_b=*/false);
  }

  if (lane < 16) {
    const float inv = 1.0f / s;  // normalize softmax after the linear combine
    v4f out;
    out[0] = acc[0] * inv;
    out[1] = acc[1] * inv;
    out[2] = acc[2] * inv;
    out[3] = acc[3] * inv;
    ((v4f*)Wfused)[col] = out;
  }
}

// ---------------------------------------------------------------------------
// Kernel 2: streaming evaluation. out[b,g] = W0 + W1*A + W2*B + W3*A*B
// Each thread: 4 gate columns (one float4 of output) x 8 batch rows, reusing
// the 64B of fused weights in registers across the 8 rows. x/out use
// non-temporal b128 accesses (streamed once); W stays resident in L2.
// ---------------------------------------------------------------------------
__global__ void lg16_eval(const float* __restrict__ x,
                          const float* __restrict__ Wfused,
                          float* __restrict__ out) {
  const int tid = blockIdx.x * blockDim.x + threadIdx.x;
  const int j = tid & (GATES / 4 - 1);  // column group: gates 4j..4j+3
  const int bg = tid >> 13;             // batch group: rows 8*bg..8*bg+7

  v4f W[4];
#pragma unroll
  for (int c = 0; c < 4; ++c) W[c] = ((const v4f*)Wfused)[4 * j + c];

  const float* xb = x + (size_t)(8 * bg) * (2 * GATES) + (size_t)(8 * j);
  float* ob = out + (size_t)(8 * bg) * GATES + (size_t)(4 * j);

#pragma unroll
  for (int i = 0; i < 8; ++i) {
    const v4f x0 = __builtin_nontemporal_load((const v4f*)xb + 0);  // A0 B0 A1 B1
    const v4f x1 = __builtin_nontemporal_load((const v4f*)xb + 1);  // A2 B2 A3 B3
    v4f r;
    r[0] = fmaf(x0[0], fmaf(W[0][3], x0[1], W[0][1]), fmaf(W[0][2], x0[1], W[0][0]));
    r[1] = fmaf(x0[2], fmaf(W[1][3], x0[3], W[1][1]), fmaf(W[1][2], x0[3], W[1][0]));
    r[2] = fmaf(x1[0], fmaf(W[2][3], x1[1], W[2][1]), fmaf(W[2][2], x1[1], W[2][0]));
    r[3] = fmaf(x1[2], fmaf(W[3][3], x1[3], W[3][1]), fmaf(W[3][2], x1[3], W[3][0]));
    __builtin_nontemporal_store(r, (v4f*)ob);
    xb += 2 * GATES;
    ob += GATES;
  }
}

extern "C" void kernel_launch(void* const* d_in, const int* in_sizes, int n_in,
                              void* d_out, int out_size, void* d_ws, size_t ws_size,
                              hipStream_t stream) {
  const float* x = (const float*)d_in[0];  // (512, 65536) f32
  const float* w = (const float*)d_in[1];  // (16, 32768) f32
  float* out = (float*)d_out;              // (512, 32768) f32
  float* Wfused = (float*)d_ws;            // 32768 * float4 = 512 KB scratch

  // 32768 columns / (16 cols/wave * 8 waves/block) = 256 blocks
  lg16_prep_weights<<<256, 256, 0, stream>>>(w, Wfused);

  // (32768/4 col-groups) * (512/8 batch-groups) = 524288 threads
  lg16_eval<<<(GATES / 4) * (BATCH / 8) / 256, 256, 0, stream>>>(x, Wfused, out);
}